// Forecaster_12214886990736
// MI455X (gfx1250) — compile-verified
//
#include <hip/hip_runtime.h>
#include <math.h>

#define HIDDEN     2048
#define H4         8192
#define INPUT_SIZE 350

typedef __attribute__((ext_vector_type(2))) float v2f;
typedef __attribute__((ext_vector_type(8))) float v8f;

__device__ __forceinline__ float sigmoidf_(float x) { return 1.0f / (1.0f + expf(-x)); }

// ---------------------------------------------------------------------------
// K1: feature preparation (350 floats) + preload layer-1 gate accumulator with
//     b_ih1 + b_hh1 (the W_hh1 @ h0 term is exactly zero since h0 == 0).
// ---------------------------------------------------------------------------
__global__ void prep_kernel(const float* __restrict__ x,       // (133,2)
                            const float* __restrict__ b_ih1,
                            const float* __restrict__ b_hh1,
                            float* __restrict__ feat,          // 352 floats
                            float* __restrict__ g1)            // 8192 floats
{
  const int tid = threadIdx.x;

  // hand bounding boxes (cheap, every thread computes them)
  float lminx = x[91 * 2 + 0], lmaxx = lminx, lminy = x[91 * 2 + 1], lmaxy = lminy;
  float rminx = x[112 * 2 + 0], rmaxx = rminx, rminy = x[112 * 2 + 1], rmaxy = rminy;
  for (int q = 1; q < 21; ++q) {
    float lx = x[(91 + q) * 2 + 0],  ly = x[(91 + q) * 2 + 1];
    float rx = x[(112 + q) * 2 + 0], ry = x[(112 + q) * 2 + 1];
    lminx = fminf(lminx, lx); lmaxx = fmaxf(lmaxx, lx);
    lminy = fminf(lminy, ly); lmaxy = fmaxf(lmaxy, ly);
    rminx = fminf(rminx, rx); rmaxx = fmaxf(rmaxx, rx);
    rminy = fminf(rminy, ry); rmaxy = fmaxf(rmaxy, ry);
  }
  const float wl = lmaxx - lminx, hl = lmaxy - lminy;
  const float wr = rmaxx - rminx, hr = rmaxy - rminy;
  float dlx = 1.f, dly = 1.f, drx = 1.f, dry = 1.f;
  if (wl != 0.f && hl != 0.f) { dlx = wl; dly = hl; }
  if (wr != 0.f && hr != 0.f) { drx = wr; dry = hr; }

  if (tid < INPUT_SIZE) {
    const int p = tid >> 1, c = tid & 1;
    const float kp0 = x[c];
    float val;
    if      (p <  17) val = x[p * 2 + c] - kp0;                           // body
    else if (p <  23) val = x[p * 2 + c];                                 // feet
    else if (p <  91) val = x[p * 2 + c] - x[53 * 2 + c];                 // face
    else if (p < 112) val = (x[p * 2 + c] - x[100 * 2 + c]) / (c ? dly : dlx); // left
    else if (p < 133) val = (x[p * 2 + c] - x[121 * 2 + c]) / (c ? dry : drx); // right
    else              val = x[(p - 42) * 2 + c] - kp0;                    // chin2l / chin2r
    feat[tid] = val;
  }
  if (tid >= INPUT_SIZE && tid < 352) feat[tid] = 0.f;

  for (int r = tid; r < H4; r += blockDim.x) g1[r] = b_ih1[r] + b_hh1[r];
}

// ---------------------------------------------------------------------------
// K2: layer-0 LSTM, fused. One block (3 waves) per hidden unit j; wave w
//     computes the dot for gate {i, g, o} (f-gate skipped: c0 == 0).
// ---------------------------------------------------------------------------
__global__ void lstm0_kernel(const float* __restrict__ W,      // (8192, 350)
                             const float* __restrict__ feat,
                             const float* __restrict__ b_ih,
                             const float* __restrict__ b_hh,
                             float* __restrict__ h1)           // 2048
{
  __shared__ float sf[INPUT_SIZE];
  __shared__ float sd[3];
  for (int t = threadIdx.x; t < INPUT_SIZE; t += 96) sf[t] = feat[t];
  __syncthreads();

  const int w = threadIdx.x >> 5, lane = threadIdx.x & 31;
  const int gate = (w == 0) ? 0 : (w + 1);        // 0->i, 1->g, 2->o
  const int row  = gate * HIDDEN + blockIdx.x;
  const float* Wr = W + (size_t)row * INPUT_SIZE;
  float s = 0.f;
  for (int k = lane; k < INPUT_SIZE; k += 32) s += Wr[k] * sf[k];
  #pragma unroll
  for (int off = 16; off > 0; off >>= 1) s += __shfl_xor(s, off, 32);
  if (lane == 0) sd[w] = s + b_ih[row] + b_hh[row];
  __syncthreads();
  if (threadIdx.x == 0) {
    const float c = sigmoidf_(sd[0]) * tanhf(sd[1]);
    h1[blockIdx.x] = sigmoidf_(sd[2]) * tanhf(c);
  }
}

// ---------------------------------------------------------------------------
// WMMA GEMV tile: full-K dot products for 16 consecutive rows of W using
// V_WMMA_F32_16X16X4_F32.  A = W tile (16x4 per step, lane L holds row L%16,
// K = vgpr + 2*(L>=16) per the ISA 32-bit A layout).  B = vector chunk
// replicated across all 16 N columns (same K mapping, mirrored).  C
// accumulates; after the loop D[m][n] = dot(W_row_m, v_chunk) for every n.
// EXEC must be all ones here: no divergence inside.
// ---------------------------------------------------------------------------
__device__ __forceinline__ v8f gemv_tile_wmma(const float* __restrict__ Wlane,
                                              const float* __restrict__ svLane,
                                              int KC)
{
  v8f acc = {0.f, 0.f, 0.f, 0.f, 0.f, 0.f, 0.f, 0.f};
  #pragma unroll 8
  for (int k = 0; k < KC; k += 4) {
    v2f a = *(const v2f*)(Wlane + k);
    v2f b = *(const v2f*)(svLane + k);
    acc = __builtin_amdgcn_wmma_f32_16x16x4_f32(false, a, false, b,
                                                (short)0, acc, false, false);
  }
  return acc;
}

// ---------------------------------------------------------------------------
// K3: layer-1 gate GEMV partials via WMMA. Grid (128 j-groups, 4 K-chunks),
//     3 waves per block = one per live gate. Partial dots atomically added
//     into the bias-preloaded g1 buffer.
// ---------------------------------------------------------------------------
__global__ void lstm1_wmma_kernel(const float* __restrict__ W,   // (8192, 2048)
                                  const float* __restrict__ v,   // h1, 2048
                                  float* __restrict__ g1)        // 8192
{
  __shared__ float sv[512];
  __shared__ float sdots[3][16];
  const int kBase = blockIdx.y * 512;
  for (int t = threadIdx.x; t < 512; t += 96) sv[t] = v[kBase + t];
  __syncthreads();

  const int w      = threadIdx.x >> 5;
  const int lane   = threadIdx.x & 31;
  const int laneLo = lane & 15;
  const int laneHi = lane >> 4;
  const int gate   = (w == 0) ? 0 : (w + 1);       // i, g, o row blocks
  const int row    = gate * HIDDEN + blockIdx.x * 16 + laneLo;

  v8f acc = gemv_tile_wmma(W + (size_t)row * HIDDEN + kBase + 2 * laneHi,
                           sv + 2 * laneHi, 512);

  // D layout: vgpr r -> M = r + 8*laneHi (lanes 0 and 16 carry all 16 rows)
  if (laneLo == 0) {
    #pragma unroll
    for (int r = 0; r < 8; ++r) sdots[w][(laneHi << 3) + r] = acc[r];
  }
  __syncthreads();
  if (threadIdx.x < 16) {
    const int j = blockIdx.x * 16 + threadIdx.x;
    atomicAdd(&g1[j],              sdots[0][threadIdx.x]);
    atomicAdd(&g1[2 * HIDDEN + j], sdots[1][threadIdx.x]);
    atomicAdd(&g1[3 * HIDDEN + j], sdots[2][threadIdx.x]);
  }
}

// ---------------------------------------------------------------------------
// K4: layer-1 gate activation -> h2, and initialize d_out with fc_b
//     (fc partials are atomically accumulated on top by K5).
// ---------------------------------------------------------------------------
__global__ void act1_kernel(const float* __restrict__ g1,
                            const float* __restrict__ fc_b,
                            float* __restrict__ h2,
                            float* __restrict__ out)
{
  const int t = blockIdx.x * blockDim.x + threadIdx.x;
  if (t < HIDDEN) {
    const float c = sigmoidf_(g1[t]) * tanhf(g1[2 * HIDDEN + t]);
    h2[t] = sigmoidf_(g1[3 * HIDDEN + t]) * tanhf(c);
  }
  if (t < INPUT_SIZE) out[t] = fc_b[t];
}

// ---------------------------------------------------------------------------
// K5: fc GEMV partials via WMMA. Grid (22 row-groups covering 352 >= 350,
//     4 K-chunks), one wave per block. Row index clamped for loads; stores
//     guarded to r < 350.
// ---------------------------------------------------------------------------
__global__ void fc_wmma_kernel(const float* __restrict__ W,     // (350, 2048)
                               const float* __restrict__ v,     // h2
                               float* __restrict__ out)         // 350
{
  __shared__ float sv[512];
  __shared__ float sdots[16];
  const int kBase = blockIdx.y * 512;
  for (int t = threadIdx.x; t < 512; t += 32) sv[t] = v[kBase + t];
  __syncthreads();

  const int lane   = threadIdx.x;
  const int laneLo = lane & 15;
  const int laneHi = lane >> 4;
  int row = blockIdx.x * 16 + laneLo;
  if (row > INPUT_SIZE - 1) row = INPUT_SIZE - 1;   // clamp (results discarded)

  v8f acc = gemv_tile_wmma(W + (size_t)row * HIDDEN + kBase + 2 * laneHi,
                           sv + 2 * laneHi, 512);

  if (laneLo == 0) {
    #pragma unroll
    for (int r = 0; r < 8; ++r) sdots[(laneHi << 3) + r] = acc[r];
  }
  __syncthreads();
  if (lane < 16) {
    const int r = blockIdx.x * 16 + lane;
    if (r < INPUT_SIZE) atomicAdd(&out[r], sdots[lane]);
  }
}

// ---------------------------------------------------------------------------
extern "C" void kernel_launch(void* const* d_in, const int* in_sizes, int n_in,
                              void* d_out, int out_size, void* d_ws, size_t ws_size,
                              hipStream_t stream)
{
  (void)in_sizes; (void)n_in; (void)out_size; (void)ws_size;

  const float* x     = (const float*)d_in[0];
  const float* W_ih0 = (const float*)d_in[1];
  // d_in[2] = W_hh0: unused (h0 == 0)
  const float* b_ih0 = (const float*)d_in[3];
  const float* b_hh0 = (const float*)d_in[4];
  const float* W_ih1 = (const float*)d_in[5];
  // d_in[6] = W_hh1: unused (h == 0)
  const float* b_ih1 = (const float*)d_in[7];
  const float* b_hh1 = (const float*)d_in[8];
  const float* fc_w  = (const float*)d_in[9];
  const float* fc_b  = (const float*)d_in[10];
  float* out = (float*)d_out;

  float* ws   = (float*)d_ws;
  float* feat = ws;            // 352
  float* g1   = ws + 512;      // 8192
  float* h1   = ws + 512 + H4;           // 2048
  float* h2   = ws + 512 + H4 + HIDDEN;  // 2048

  prep_kernel      <<<1, 512, 0, stream>>>(x, b_ih1, b_hh1, feat, g1);
  lstm0_kernel     <<<HIDDEN, 96, 0, stream>>>(W_ih0, feat, b_ih0, b_hh0, h1);
  lstm1_wmma_kernel<<<dim3(HIDDEN / 16, 4), 96, 0, stream>>>(W_ih1, h1, g1);
  act1_kernel      <<<(HIDDEN + 255) / 256, 256, 0, stream>>>(g1, fc_b, h2, out);
  fc_wmma_kernel   <<<dim3(22, 4), 32, 0, stream>>>(fc_w, h2, out);
}